// AutoEncoder_39195871543876
// MI455X (gfx1250) — compile-verified
//
#include <hip/hip_runtime.h>

// Problem constants (match reference setup: B=16, N=1e6, xsize=128)
#define XS    128
#define VOX   (XS * XS * XS)   // 2,097,152 voxels per batch
#define BATCH 16

typedef float v2f __attribute__((ext_vector_type(2)));
typedef float v8f __attribute__((ext_vector_type(8)));

// ---------------------------------------------------------------------------
// Kernel 0: zero the workspace (grid + accumulators). float4 stores.
// ---------------------------------------------------------------------------
__global__ void tvgrid_zero_kernel(float4* __restrict__ ws, long n4) {
    long i = (long)blockIdx.x * blockDim.x + threadIdx.x;
    long stride = (long)gridDim.x * blockDim.x;
    float4 z = make_float4(0.f, 0.f, 0.f, 0.f);
    for (; i < n4; i += stride) ws[i] = z;
}

// ---------------------------------------------------------------------------
// Kernel 1: scatter-add. One thread per point; lin index computed once,
// then 16 non-returning global_atomic_add_f32 (fire-and-forget, STOREcnt).
// values reads are coalesced across lanes for each fixed b.
// ---------------------------------------------------------------------------
__global__ __launch_bounds__(256)
void tvgrid_scatter_kernel(const float* __restrict__ values,
                           const int* __restrict__ indices,
                           float* __restrict__ grid, int n) {
    int p = blockIdx.x * blockDim.x + threadIdx.x;
    if (p >= n) return;
    int i0 = indices[3 * p + 0];
    int i1 = indices[3 * p + 1];
    int i2 = indices[3 * p + 2];
    int lin = (i0 * XS + i1) * XS + i2;
#pragma unroll
    for (int b = 0; b < BATCH; ++b) {
        atomicAdd(grid + (size_t)b * VOX + lin, values[(size_t)b * n + p]);
    }
}

// ---------------------------------------------------------------------------
// Exact f32 wave(32)-wide sum via V_WMMA_F32_16X16X4_F32 with B = ones.
// A layout (16x4 f32): VGPR0 = K0 (lanes 0-15) / K2 (lanes 16-31), VGPR1 = K1/K3.
// With a = {p, 0}:  D[m][n] = p_m + p_{m+16}  (same for every column n).
// Lane l holds C rows {g + 8*(l>>4)} at col l&15 -> per-lane sum of the 8
// accumulator VGPRs gives the half-total; one shfl_xor(16) add -> full total.
// B = all ones is layout-agnostic. EXEC is all-1s here (no divergence).
// ---------------------------------------------------------------------------
__device__ __forceinline__ float wave_sum_wmma(float v) {
    v2f a;   a[0] = v;   a[1] = 0.0f;
    v2f one; one[0] = 1.0f; one[1] = 1.0f;
    v8f c = {};
    c = __builtin_amdgcn_wmma_f32_16x16x4_f32(
            /*neg_a=*/false, a, /*neg_b=*/false, one,
            /*c_mod=*/(short)0, c, /*reuse_a=*/false, /*reuse_b=*/false);
    float s = ((c[0] + c[1]) + (c[2] + c[3])) + ((c[4] + c[5]) + (c[6] + c[7]));
    s += __shfl_xor(s, 16, 32);
    return s;
}

// ---------------------------------------------------------------------------
// Kernel 2: neighbor-difference TV + MSE reduction.
// 1024 blocks, 256 threads, each block owns 32768 contiguous voxels of ONE
// batch (64 blocks per batch). Grid fits in the 192MB L2, so the 4x read
// amplification stays on-chip. Hierarchy: thread-acc -> WMMA wave sum ->
// LDS block sum -> 1 atomicAdd per block per quantity (low contention).
// ---------------------------------------------------------------------------
__global__ __launch_bounds__(256)
void tvgrid_diff_kernel(const float* __restrict__ grid, float* __restrict__ acc) {
    const int EPB = 32768;            // elements per block (VOX / 64)
    const int ITERS = EPB / 256;      // 128
    int blk = blockIdx.x;
    int b = blk >> 6;                 // 64 blocks per batch
    const float* gb = grid + (size_t)b * VOX;
    int base = (blk & 63) * EPB + (int)threadIdx.x;

    float tv = 0.f, ms = 0.f;
#pragma unroll 4
    for (int k = 0; k < ITERS; ++k) {
        int e = base + k * 256;
        int x = e & (XS - 1);
        int y = (e >> 7) & (XS - 1);
        int z = e >> 14;
        float c = gb[e];
        if (x < XS - 1) { float d = gb[e + 1]       - c; tv += fabsf(d); ms += d * d; }
        if (y < XS - 1) { float d = gb[e + XS]      - c; tv += fabsf(d); ms += d * d; }
        if (z < XS - 1) { float d = gb[e + XS * XS] - c; tv += fabsf(d); ms += d * d; }
    }

    float wtv = wave_sum_wmma(tv);
    float wms = wave_sum_wmma(ms);

    __shared__ float stv[8];
    __shared__ float sms[8];
    int lane = threadIdx.x & 31;
    int wave = threadIdx.x >> 5;
    if (lane == 0) { stv[wave] = wtv; sms[wave] = wms; }
    __syncthreads();
    if (threadIdx.x == 0) {
        float t = 0.f, m = 0.f;
#pragma unroll
        for (int w = 0; w < 8; ++w) { t += stv[w]; m += sms[w]; }
        atomicAdd(acc + b, t);
        atomicAdd(acc + BATCH + b, m);
    }
}

// ---------------------------------------------------------------------------
// Kernel 3: scale accumulators into d_out: tv / xs^3, mse / (2*xs^2 - 2*xs)
// ---------------------------------------------------------------------------
__global__ void tvgrid_finalize_kernel(const float* __restrict__ acc,
                                       float* __restrict__ out) {
    int i = threadIdx.x;
    if (i < BATCH)           out[i] = acc[i] * (1.0f / (float)VOX);
    else if (i < 2 * BATCH)  out[i] = acc[i] * (1.0f / (float)(2 * XS * XS - 2 * XS));
}

// ---------------------------------------------------------------------------
// Workspace layout: [0, 16*VOX) floats = grid; [16*VOX, 16*VOX+32) = acc.
// Requires ws_size >= (16*VOX + 32) * 4 bytes (~128 MB).
// ---------------------------------------------------------------------------
extern "C" void kernel_launch(void* const* d_in, const int* in_sizes, int n_in,
                              void* d_out, int out_size, void* d_ws, size_t ws_size,
                              hipStream_t stream) {
    const float* values  = (const float*)d_in[0];   // (B, N) f32
    const int*   indices = (const int*)d_in[1];     // (N, 3) i32
    int n = in_sizes[1] / 3;                        // N points

    float* grid = (float*)d_ws;
    float* acc  = grid + (size_t)BATCH * VOX;

    long n4 = ((long)BATCH * VOX + 32) / 4;         // floats -> float4 count
    tvgrid_zero_kernel<<<8192, 256, 0, stream>>>((float4*)d_ws, n4);

    tvgrid_scatter_kernel<<<(n + 255) / 256, 256, 0, stream>>>(values, indices, grid, n);

    tvgrid_diff_kernel<<<BATCH * 64, 256, 0, stream>>>(grid, acc);

    tvgrid_finalize_kernel<<<1, 2 * BATCH, 0, stream>>>(acc, (float*)d_out);
}